// ISTAAttention_44143673868412
// MI455X (gfx1250) — compile-verified
//
#include <hip/hip_runtime.h>
#include <hip/hip_bf16.h>

// ISTA attention for MI455X (gfx1250, wave32, WMMA).
// Flash-style two-pass attention (no 402MB attn materialization), algebraic
// fusion grad2-grad1 = attn^T @ (v - x1), all matmuls on V_WMMA_F32_16X16X4_F32
// for f32 fidelity. k/v tiles staged to LDS with GLOBAL_LOAD_ASYNC_TO_LDS_B128
// (double-buffered, ASYNCcnt-fenced) and shared by the 4 waves of each block.

typedef __attribute__((ext_vector_type(2))) float v2f;
typedef __attribute__((ext_vector_type(8))) float v8f;
typedef __attribute__((ext_vector_type(4))) int   v4i;

#define B_  2
#define N_  2048
#define C_  768
#define H_  12
#define D_  64
#define N16 (N_ / 16)
#define QKV_ROW (3 * C_)   // 2304 floats per token row in the qkv buffer
#define TPITCH 68          // LDS row pitch (floats): 16B aligned, no bank conflicts

__device__ __forceinline__ v8f wmma4(v2f a, v2f b, v8f c) {
  // D = A(16x4,f32) * B(4x16,f32) + C(16x16,f32)
  return __builtin_amdgcn_wmma_f32_16x16x4_f32(
      /*neg_a=*/false, a, /*neg_b=*/false, b,
      /*c_mod=*/(short)0, c, /*reuse_a=*/false, /*reuse_b=*/false);
}

// ---- CDNA5 async global->LDS copy (ASYNCcnt-tracked DMA) -------------------
__device__ __forceinline__ void async_copy_b128(const float* g, float* l) {
#if defined(__has_builtin) && __has_builtin(__builtin_amdgcn_global_load_async_to_lds_b128)
  __builtin_amdgcn_global_load_async_to_lds_b128(
      (__attribute__((address_space(1))) v4i*)g,
      (__attribute__((address_space(3))) v4i*)l, 0, 0);
#else
  const unsigned loff =
      (unsigned)(unsigned long long)(__attribute__((address_space(3))) float*)l;
  asm volatile("global_load_async_to_lds_b128 %0, %1, off"
               :: "v"(loff), "v"(g) : "memory");
#endif
}

__device__ __forceinline__ void wait_asynccnt0() {
#if defined(__has_builtin) && __has_builtin(__builtin_amdgcn_s_wait_asynccnt)
  __builtin_amdgcn_s_wait_asynccnt(0);
#else
  asm volatile("s_wait_asynccnt 0" ::: "memory");
#endif
}

// ---------------------------------------------------------------------------
// GEMM: Out[m,n] = sum_k A[m,k] * W[n,k] (+ bias[n]); computes A @ W^T.
// One wave per 16x16 tile; K-loop of f32 16x16x4 WMMAs; contiguous float2
// per lane for both operands per the 32-bit WMMA VGPR layouts.
// ---------------------------------------------------------------------------
__global__ void __launch_bounds__(128)
gemm_xWT(const float* __restrict__ A, const float* __restrict__ W,
         const float* __restrict__ bias, float* __restrict__ Out,
         int M, int Nn, int K) {
  const int lane = threadIdx.x & 31;
  const int wid  = threadIdx.x >> 5;
  const int tile = blockIdx.x * 4 + wid;
  const int tilesN = Nn >> 4;
  const int tM = tile / tilesN;
  const int tN = tile % tilesN;
  const int l15  = lane & 15;
  const int kOff = (lane >> 4) << 1;

  const float* arow = A + (size_t)(tM * 16 + l15) * K + kOff;
  const float* wrow = W + (size_t)(tN * 16 + l15) * K + kOff;

  v8f acc = {};
#pragma unroll 8
  for (int k0 = 0; k0 < K; k0 += 4) {
    v2f a = *(const v2f*)(arow + k0);
    v2f b = *(const v2f*)(wrow + k0);
    acc = wmma4(a, b, acc);
  }

  const int   col = tN * 16 + l15;
  const float bv  = bias ? bias[col] : 0.0f;
  const int   rBase = tM * 16 + ((lane >> 4) << 3);
#pragma unroll
  for (int r = 0; r < 8; ++r)
    Out[(size_t)(rBase + r) * Nn + col] = acc[r] + bv;
}

// ---------------------------------------------------------------------------
// In-place L2 normalize q and k rows of the qkv buffer.
// ---------------------------------------------------------------------------
__global__ void l2norm_qk(float* __restrict__ qkv) {
  const int idx = blockIdx.x * blockDim.x + threadIdx.x;  // B*N*2*H threads
  const int h   = idx % H_;
  const int t   = idx / H_;
  const int sel = t & 1;       // 0 = q, 1 = k
  const int bn  = t >> 1;      // token index over B*N
  float* row = qkv + (size_t)bn * QKV_ROW + sel * C_ + h * D_;
  float s = 0.0f;
#pragma unroll 8
  for (int d = 0; d < D_; ++d) { float v = row[d]; s += v * v; }
  const float inv = 1.0f / fmaxf(sqrtf(s), 1e-12f);
#pragma unroll 8
  for (int d = 0; d < D_; ++d) row[d] *= inv;
}

// ---------------------------------------------------------------------------
// Flash-style pass. One block = 4 waves on 4 consecutive 16-row i-tiles of the
// SAME (b,h); the block streams all 128 key tiles with k/v staged into
// double-buffered LDS via async b128 DMA and shared by all 4 waves:
//   S = (A_i . B_j^T) * scale           (16x16x64 via 16 WMMAs, B from LDS)
//   acc[d-tile] += S @ V_j              (S transposed via LDS, V from LDS)
// mode 0 (pass A): A=q, B=k, V=v   -> y = v - x1            ([B,H,N,D])
// mode 1 (pass B): A=k, B=q, V=y   -> relu(v + 0.1*g - .05) ([B,N,C])
// ---------------------------------------------------------------------------
__global__ void __launch_bounds__(128)
ista_attn_pass(const float* __restrict__ qkv, float* __restrict__ y,
               float* __restrict__ outPre, int selA, int selB, int mode) {
  __shared__ __align__(16) float kT[2][16 * TPITCH];   // double-buffered key tile
  __shared__ __align__(16) float vT[2][16 * TPITCH];   // double-buffered value tile
  __shared__ __align__(16) float sT[4][16 * 17];       // per-wave S transpose tile

  const int tid  = threadIdx.x;
  const int lane = tid & 31;
  const int wid  = tid >> 5;
  const int bh   = blockIdx.x / (N16 / 4);
  const int iT   = (blockIdx.x % (N16 / 4)) * 4 + wid;
  const int h    = bh % H_;
  const int b    = bh / H_;
  const int l15  = lane & 15;
  const int kOff = (lane >> 4) << 1;
  const int rB   = (lane >> 4) << 3;

  // --- staging sources (shared across the block) -------------------------
  const float* kSrc = qkv + (size_t)b * N_ * QKV_ROW + selB * C_ + h * D_;
  const float* vSrc;
  size_t vStride;
  if (mode == 0) { vSrc = qkv + (size_t)b * N_ * QKV_ROW + 2 * C_ + h * D_; vStride = QKV_ROW; }
  else           { vSrc = y + (size_t)(b * H_ + h) * N_ * D_;               vStride = D_; }

  const int srow = tid >> 3;        // 0..15 : tile row this thread stages
  const int scol = (tid & 7) * 8;   // 0,8,..,56 : 8-float chunk

  auto stage = [&](int jT, int buf) {
    const float* kg = kSrc + (size_t)(jT * 16 + srow) * QKV_ROW + scol;
    const float* vg = vSrc + (size_t)(jT * 16 + srow) * vStride + scol;
    float* kl = &kT[buf][srow * TPITCH + scol];
    float* vl = &vT[buf][srow * TPITCH + scol];
    async_copy_b128(kg,     kl);
    async_copy_b128(kg + 4, kl + 4);
    async_copy_b128(vg,     vl);
    async_copy_b128(vg + 4, vl + 4);
  };

  // --- hoist this wave's A-operand tile (16x64) into registers ----------
  const float* qb   = qkv + (size_t)b * N_ * QKV_ROW + selA * C_ + h * D_;
  const float* qrow = qb + (size_t)(iT * 16 + l15) * QKV_ROW + kOff;
  v2f qa[16];
#pragma unroll
  for (int i = 0; i < 16; ++i) qa[i] = *(const v2f*)(qrow + i * 4);

  float* sl = &sT[wid][0];
  const float scale = 0.125f;  // D^-0.5

  v8f zero = {};
  v8f acc[4];
#pragma unroll
  for (int t = 0; t < 4; ++t) acc[t] = zero;

  stage(0, 0);
  for (int jT = 0; jT < N16; ++jT) {
    const int cur = jT & 1;
    wait_asynccnt0();          // our staged tile landed in LDS
    __syncthreads();           // everyone's chunks visible; prev buffer free
    if (jT + 1 < N16) stage(jT + 1, cur ^ 1);

    // --- S = A_i . B_j^T (B operand from LDS) --------------------------
    v8f s = zero;
    const float* kl = &kT[cur][0];
#pragma unroll
    for (int kd4 = 0; kd4 < 16; ++kd4) {
      v2f bm = *(const v2f*)(kl + l15 * TPITCH + kd4 * 4 + kOff);
      s = wmma4(qa[kd4], bm, s);
    }

    // --- transpose S through LDS (C-layout -> A-operand layout) --------
#pragma unroll
    for (int r = 0; r < 8; ++r)
      sl[(rB + r) * 17 + l15] = s[r] * scale;
    asm volatile("s_wait_dscnt 0" ::: "memory");

    // --- acc += S @ V tile (V operand from LDS) ------------------------
    const float* vl = &vT[cur][0];
#pragma unroll
    for (int kk = 0; kk < 16; kk += 4) {
      v2f a;
      a.x = sl[l15 * 17 + kk + kOff];
      a.y = sl[l15 * 17 + kk + kOff + 1];
      const float* v0p = vl + (kk + kOff) * TPITCH + l15;
      const float* v1p = v0p + TPITCH;
#pragma unroll
      for (int t = 0; t < 4; ++t) {
        v2f bm;
        bm.x = v0p[t * 16];
        bm.y = v1p[t * 16];
        acc[t] = wmma4(a, bm, acc[t]);
      }
    }
  }

  // --- epilogue ----------------------------------------------------------
  const float* vq = qkv + (size_t)b * N_ * QKV_ROW + 2 * C_ + h * D_;
  if (mode == 0) {
    float* yo = y + (size_t)(b * H_ + h) * N_ * D_;
#pragma unroll
    for (int t = 0; t < 4; ++t)
#pragma unroll
      for (int r = 0; r < 8; ++r) {
        const int m = iT * 16 + rB + r;
        const int d = t * 16 + l15;
        yo[(size_t)m * D_ + d] = vq[(size_t)m * QKV_ROW + d] - acc[t][r];
      }
  } else {
    float* op = outPre + (size_t)b * N_ * C_ + h * D_;
#pragma unroll
    for (int t = 0; t < 4; ++t)
#pragma unroll
      for (int r = 0; r < 8; ++r) {
        const int m = iT * 16 + rB + r;
        const int d = t * 16 + l15;
        const float v = vq[(size_t)m * QKV_ROW + d];
        op[(size_t)m * C_ + d] = fmaxf(v + 0.1f * acc[t][r] - 0.05f, 0.0f);
      }
  }
}

// ---------------------------------------------------------------------------
extern "C" void kernel_launch(void* const* d_in, const int* in_sizes, int n_in,
                              void* d_out, int out_size, void* d_ws, size_t ws_size,
                              hipStream_t stream) {
  const float* x     = (const float*)d_in[0];
  const float* Wqkv  = (const float*)d_in[1];
  const float* Wproj = (const float*)d_in[2];
  const float* bproj = (const float*)d_in[3];
  float* out = (float*)d_out;

  // Workspace: qkv [B,N,3C] | y [B,H,N,D] | outPre [B,N,C]  (~63 MB total)
  float* qkv = (float*)d_ws;
  float* yb  = qkv + (size_t)B_ * N_ * 3 * C_;
  float* op  = yb  + (size_t)B_ * H_ * N_ * D_;

  const int M = B_ * N_;  // 4096

  // 1) qkv = x @ Wqkv^T    (M=4096, N=2304, K=768)
  {
    const int tiles = (M / 16) * ((3 * C_) / 16);  // 36864
    gemm_xWT<<<tiles / 4, 128, 0, stream>>>(x, Wqkv, nullptr, qkv, M, 3 * C_, C_);
  }
  // 2) L2-normalize q and k
  l2norm_qk<<<(B_ * N_ * 2 * H_) / 256, 256, 0, stream>>>(qkv);
  // 3) pass A: y = v - (scale*q k^T) @ v
  {
    const int blocks = B_ * H_ * (N16 / 4);  // 768
    ista_attn_pass<<<blocks, 128, 0, stream>>>(qkv, yb, op, 0, 1, 0);
  }
  // 4) pass B: outPre = relu(v + 0.1 * (scale*k q^T) @ y - 0.05)
  {
    const int blocks = B_ * H_ * (N16 / 4);
    ista_attn_pass<<<blocks, 128, 0, stream>>>(qkv, yb, op, 1, 0, 1);
  }
  // 5) out = outPre @ Wproj^T + bproj   (M=4096, N=768, K=768)
  {
    const int tiles = (M / 16) * (C_ / 16);  // 12288
    gemm_xWT<<<tiles / 4, 128, 0, stream>>>(op, Wproj, bproj, out, M, C_, C_);
  }
}